// Model_77635828842803
// MI455X (gfx1250) — compile-verified
//
#include <hip/hip_runtime.h>
#include <math.h>

typedef __attribute__((ext_vector_type(2))) float v2f;
typedef __attribute__((ext_vector_type(8))) float v8f;

#define NN 4096
#define NH 512
#define SS 10

// Toggle the gfx1250 async global->LDS staging path (ASYNCcnt-tracked).
#define USE_ASYNC_LDS 1

// 16B global -> LDS direct copy, no VGPR round-trip. VDST operand is the
// 32-bit LDS byte address (= low 32 bits of the generic pointer, since the
// LDS aperture keeps the LDS offset in addr[31:0]).
__device__ __forceinline__ void async_copy_b128(const float* gsrc, float* ldst) {
#if USE_ASYNC_LDS
  unsigned l = (unsigned)(unsigned long long)ldst;
  unsigned long long g = (unsigned long long)gsrc;
  asm volatile("global_load_async_to_lds_b128 %0, %1, off"
               :: "v"(l), "v"(g) : "memory");
#else
  *(float4*)ldst = *(const float4*)gsrc;
#endif
}

__device__ __forceinline__ void wait_async_zero() {
#if USE_ASYNC_LDS
  asm volatile("s_wait_asynccnt 0x0" ::: "memory");
#endif
}

// ============================================================================
// WMMA f32 GEMM:  C[M,N] = A[M,K] * B   (V_WMMA_F32_16X16X4_F32)
//   TRANSB==1 : B stored [N,K] row-major (torch weight / symmetric Gram)
//   TRANSB==0 : B stored [K,N] row-major (adj @ fts)
//   EPI==0 : C = acc
//   EPI==1 : C = prelu(acc + bias[n], *slope_p)
//   EPI==2 : C = exp(2 * acc * acc)
// Block tile 128x64, K-tile 16, 256 threads = 8 waves (4 in M x 2 in N);
// each wave owns a 32x32 C tile = 4 WMMA accumulators (best LDS economy:
// 4 LDS v2f reads feed 4 WMMAs per k-step).
// Requires M % 128 == 0, N % 64 == 0, K % 16 == 0 (true: 4096/512).
// ============================================================================
template <int TRANSB, int EPI>
__global__ __launch_bounds__(256) void gemm_wmma_f32(
    const float* __restrict__ A, const float* __restrict__ B,
    float* __restrict__ C, int M, int N, int K,
    const float* __restrict__ bias, const float* __restrict__ slope_p)
{
  constexpr int KT = 16;
  __shared__ float As[128][KT + 4];   // [m][k], row stride 80B (16B aligned)
  __shared__ float Bs[64][KT + 4];    // [n][k]

  const int tid = threadIdx.x;
  const int bm  = blockIdx.y * 128;
  const int bn  = blockIdx.x * 64;

  const int ldr = tid >> 2;          // 0..63 loader row
  const int ldc = (tid & 3) << 2;    // 0,4,8,12

  const int w    = tid >> 5;         // wave 0..7
  const int wm   = (w & 3) * 32;     // wave row offset (4 waves in M)
  const int wn   = (w >> 2) * 32;    // wave col offset (2 waves in N)
  const int lane = tid & 31;
  const int hlf  = lane >> 4;        // lane half per f32 WMMA layout
  const int l16  = lane & 15;

  v8f acc00 = {0.f,0.f,0.f,0.f,0.f,0.f,0.f,0.f};
  v8f acc01 = acc00, acc10 = acc00, acc11 = acc00;

  for (int k0 = 0; k0 < K; k0 += KT) {
    // ---- stage A tile (128x16) : two 16B bursts per thread ----
    async_copy_b128(A + (size_t)(bm + ldr) * K + (k0 + ldc),      &As[ldr][ldc]);
    async_copy_b128(A + (size_t)(bm + ldr + 64) * K + (k0 + ldc), &As[ldr + 64][ldc]);
    // ---- stage B tile as Bs[n][k] ----
    if (TRANSB) {
      async_copy_b128(B + (size_t)(bn + ldr) * K + (k0 + ldc), &Bs[ldr][ldc]);
    } else {
      const int n  = tid & 63;
      const int kb = (tid >> 6) << 2;
#pragma unroll
      for (int kk = 0; kk < 4; ++kk)
        Bs[n][kb + kk] = B[(size_t)(k0 + kb + kk) * N + (bn + n)];
    }
    if (k0 + KT < K)  // prefetch next A tile -> global_prefetch_b8
      __builtin_prefetch(A + (size_t)(bm + ldr) * K + (k0 + KT + ldc), 0, 1);
    wait_async_zero();
    __syncthreads();

    // ---- 4 k-steps x 4 WMMA (2x2 subtiles) ----
#pragma unroll
    for (int kk = 0; kk < KT; kk += 4) {
      const int ks = kk + 2 * hlf;   // lanes 0-15: k0..1, lanes 16-31: k2..3
      v2f a0 = *(const v2f*)&As[wm + l16][ks];
      v2f a1 = *(const v2f*)&As[wm + 16 + l16][ks];
      v2f b0 = *(const v2f*)&Bs[wn + l16][ks];
      v2f b1 = *(const v2f*)&Bs[wn + 16 + l16][ks];
      acc00 = __builtin_amdgcn_wmma_f32_16x16x4_f32(false, a0, false, b0, (short)0, acc00, false, false);
      acc01 = __builtin_amdgcn_wmma_f32_16x16x4_f32(false, a0, false, b1, (short)0, acc01, false, false);
      acc10 = __builtin_amdgcn_wmma_f32_16x16x4_f32(false, a1, false, b0, (short)0, acc10, false, false);
      acc11 = __builtin_amdgcn_wmma_f32_16x16x4_f32(false, a1, false, b1, (short)0, acc11, false, false);
    }
    __syncthreads();
  }

  const float slope = (EPI == 1) ? slope_p[0] : 0.f;
#pragma unroll
  for (int r = 0; r < 2; ++r) {
#pragma unroll
    for (int c = 0; c < 2; ++c) {
      v8f acc = (r == 0) ? (c == 0 ? acc00 : acc01) : (c == 0 ? acc10 : acc11);
      const int col = bn + wn + c * 16 + l16;
      const float bcol = (EPI == 1) ? bias[col] : 0.f;
#pragma unroll
      for (int v = 0; v < 8; ++v) {
        const int row = bm + wm + r * 16 + hlf * 8 + v;  // C layout: M=v / v+8
        float x = acc[v];
        if (EPI == 1) { x += bcol; x = (x >= 0.f) ? x : slope * x; }
        if (EPI == 2) { x = __expf(2.0f * x * x); }
        C[(size_t)row * N + col] = x;
      }
    }
  }
}

// ============================================================================
// Elementwise / reduction helpers
// ============================================================================
__global__ __launch_bounds__(256) void zadd_kernel(const float* __restrict__ a,
                                                   const float* __restrict__ b,
                                                   float* __restrict__ z, int n) {
  int i = blockIdx.x * 256 + threadIdx.x;
  if (i < n) z[i] = a[i] + b[i];
}

// c[h] = sigmoid(mean_n h[n][h])
__global__ __launch_bounds__(256) void colmean_sigmoid(const float* __restrict__ h,
                                                       float* __restrict__ c) {
  int col = blockIdx.x * 256 + threadIdx.x;
  if (col < NH) {
    float s = 0.f;
    for (int r = 0; r < NN; ++r) s += h[(size_t)r * NH + col];
    s *= (1.0f / NN);
    c[col] = 1.0f / (1.0f + __expf(-s));
  }
}

// cn[h] = max(||Z[:,h]||, 1e-12)
__global__ __launch_bounds__(256) void colnorm_kernel(const float* __restrict__ Z,
                                                      float* __restrict__ cn) {
  int col = blockIdx.x * 256 + threadIdx.x;
  if (col < NH) {
    float s = 0.f;
    for (int r = 0; r < NN; ++r) { float v = Z[(size_t)r * NH + col]; s += v * v; }
    cn[col] = fmaxf(sqrtf(s), 1e-12f);
  }
}

// Zn[n,:] = Z[n,:] / max(||Z[n,:]||, 1e-12)   (one block per row)
__global__ __launch_bounds__(256) void rownorm_normalize(const float* __restrict__ Z,
                                                         float* __restrict__ Zn) {
  __shared__ float red[256];
  int row = blockIdx.x;
  const float* zr = Z + (size_t)row * NH;
  float s = 0.f;
  for (int c = threadIdx.x; c < NH; c += 256) { float v = zr[c]; s += v * v; }
  red[threadIdx.x] = s; __syncthreads();
  for (int off = 128; off > 0; off >>= 1) {
    if (threadIdx.x < off) red[threadIdx.x] += red[threadIdx.x + off];
    __syncthreads();
  }
  float inv = 1.0f / fmaxf(sqrtf(red[0]), 1e-12f);
  for (int c = threadIdx.x; c < NH; c += 256) Zn[(size_t)row * NH + c] = zr[c] * inv;
}

// Zt[h][n] = Z[n][h] / cn[h]   (32x32 LDS tiles, coalesced both ways)
__global__ __launch_bounds__(256) void transpose_coln(const float* __restrict__ Z,
                                                      const float* __restrict__ cn,
                                                      float* __restrict__ Zt) {
  __shared__ float tile[32][33];
  int hb = blockIdx.x * 32;   // along NH
  int nb = blockIdx.y * 32;   // along NN
  int tx = threadIdx.x & 31;
  int ty = threadIdx.x >> 5;  // 0..7
  for (int i = 0; i < 32; i += 8)
    tile[ty + i][tx] = Z[(size_t)(nb + ty + i) * NH + (hb + tx)];
  __syncthreads();
  for (int i = 0; i < 32; i += 8) {
    int hh = hb + ty + i;
    float inv = 1.0f / cn[hh];
    Zt[(size_t)hh * NN + (nb + tx)] = tile[tx][ty + i] * inv;
  }
}

// y = W @ x, W [512,512] row-major; one wave32 per output row.
__global__ __launch_bounds__(256) void gemv512(const float* __restrict__ W,
                                               const float* __restrict__ x,
                                               float* __restrict__ y) {
  int row  = (blockIdx.x * 256 + threadIdx.x) >> 5;
  int lane = threadIdx.x & 31;
  if (row < NH) {
    const float* wr = W + (size_t)row * NH;
    float s = 0.f;
    for (int i = lane; i < NH; i += 32) s += wr[i] * x[i];
    for (int off = 16; off > 0; off >>= 1) s += __shfl_xor(s, off, 32);
    if (lane == 0) y[row] = s;
  }
}

// ret = [h2.u1, h1.u2, h4.u1, h3.u2] + bd  (wave32 per node, 4 dots fused)
__global__ __launch_bounds__(256) void disc_kernel(
    const float* __restrict__ h1, const float* __restrict__ h2,
    const float* __restrict__ h3, const float* __restrict__ h4,
    const float* __restrict__ u1, const float* __restrict__ u2,
    const float* __restrict__ bd, float* __restrict__ ret) {
  int node = (blockIdx.x * 256 + threadIdx.x) >> 5;
  int lane = threadIdx.x & 31;
  if (node < NN) {
    size_t base = (size_t)node * NH;
    float s1 = 0.f, s2 = 0.f, s3 = 0.f, s4 = 0.f;
    for (int i = lane; i < NH; i += 32) {
      float a = u1[i], b = u2[i];
      s1 += h2[base + i] * a;
      s2 += h1[base + i] * b;
      s3 += h4[base + i] * a;
      s4 += h3[base + i] * b;
    }
    for (int off = 16; off > 0; off >>= 1) {
      s1 += __shfl_xor(s1, off, 32); s2 += __shfl_xor(s2, off, 32);
      s3 += __shfl_xor(s3, off, 32); s4 += __shfl_xor(s4, off, 32);
    }
    if (lane == 0) {
      float bb = bd[0];
      ret[node]          = s1 + bb;
      ret[NN + node]     = s2 + bb;
      ret[2 * NN + node] = s3 + bb;
      ret[3 * NN + node] = s4 + bb;
    }
  }
}

// feat loss per column: sim_f symmetric -> column == row (coalesced).
__global__ __launch_bounds__(256) void featloss_cols(const float* __restrict__ simf,
                                                     const int* __restrict__ fidx,
                                                     float* __restrict__ fterm) {
  __shared__ float s[NH];
  int col = blockIdx.x;
  const float* rr = simf + (size_t)col * NH;
  for (int i = threadIdx.x; i < NH; i += 256) s[i] = rr[i];
  __syncthreads();
  for (int k = 2; k <= NH; k <<= 1) {
    for (int j = k >> 1; j > 0; j >>= 1) {
      for (int i = threadIdx.x; i < NH; i += 256) {
        int ixj = i ^ j;
        if (ixj > i) {
          float a = s[i], b = s[ixj];
          bool up = ((i & k) == 0);
          if ((a > b) == up) { s[i] = b; s[ixj] = a; }
        }
      }
      __syncthreads();
    }
  }
  if (threadIdx.x == 0) {
    float neg = 0.f;
    for (int t = 0; t < SS; ++t) neg += s[fidx[t]];
    float pos = rr[col];                 // diagonal, re-read from global
    fterm[col] = -__logf(pos / neg);
  }
}

// node loss per row n: pos = sum(sim_n[n,:]*albl[n,:]); sim_n symmetric so
// column n == row n for the sort; sort 4096 in LDS, pick ranks node_index.
__global__ __launch_bounds__(256) void nodeloss_rows(const float* __restrict__ simn,
                                                     const float* __restrict__ albl,
                                                     const int* __restrict__ nidx,
                                                     float* __restrict__ nterm) {
  __shared__ float s[NN];       // 16 KB of the 320 KB LDS
  __shared__ float red[256];
  int row = blockIdx.x;
  const float* sr = simn + (size_t)row * NN;
  const float* ar = albl + (size_t)row * NN;
  float p = 0.f;
  for (int i = threadIdx.x; i < NN; i += 256) { float v = sr[i]; s[i] = v; p += v * ar[i]; }
  red[threadIdx.x] = p; __syncthreads();
  for (int off = 128; off > 0; off >>= 1) {
    if (threadIdx.x < off) red[threadIdx.x] += red[threadIdx.x + off];
    __syncthreads();
  }
  for (int k = 2; k <= NN; k <<= 1) {
    for (int j = k >> 1; j > 0; j >>= 1) {
      for (int i = threadIdx.x; i < NN; i += 256) {
        int ixj = i ^ j;
        if (ixj > i) {
          float a = s[i], b = s[ixj];
          bool up = ((i & k) == 0);
          if ((a > b) == up) { s[i] = b; s[ixj] = a; }
        }
      }
      __syncthreads();
    }
  }
  if (threadIdx.x == 0) {
    float neg = 0.f;
    for (int t = 0; t < SS; ++t) neg += s[nidx[t]];
    nterm[row] = -__logf(red[0] / neg);
  }
}

// deterministic final means -> out[16384], out[16385]
__global__ __launch_bounds__(256) void finalize_losses(const float* __restrict__ fterm,
                                                       const float* __restrict__ nterm,
                                                       float* __restrict__ out) {
  __shared__ float red[256];
  int t = threadIdx.x;
  float s = 0.f;
  for (int i = t; i < NH; i += 256) s += fterm[i];
  red[t] = s; __syncthreads();
  for (int off = 128; off > 0; off >>= 1) { if (t < off) red[t] += red[t + off]; __syncthreads(); }
  float fl = red[0] * (1.0f / NH);
  __syncthreads();
  float s2 = 0.f;
  for (int i = t; i < NN; i += 256) s2 += nterm[i];
  red[t] = s2; __syncthreads();
  for (int off = 128; off > 0; off >>= 1) { if (t < off) red[t] += red[t + off]; __syncthreads(); }
  if (t == 0) { out[4 * NN] = fl; out[4 * NN + 1] = red[0] * (1.0f / NN); }
}

// ============================================================================
extern "C" void kernel_launch(void* const* d_in, const int* in_sizes, int n_in,
                              void* d_out, int out_size, void* d_ws, size_t ws_size,
                              hipStream_t stream) {
  (void)in_sizes; (void)n_in; (void)out_size; (void)ws_size;
  const float* seq1 = (const float*)d_in[0];
  const float* seq2 = (const float*)d_in[1];
  const float* adj  = (const float*)d_in[2];
  const float* diff = (const float*)d_in[3];
  const float* albl = (const float*)d_in[4];
  const int*   fidx = (const int*)d_in[5];
  const int*   nidx = (const int*)d_in[6];
  const float* W1 = (const float*)d_in[7];
  const float* b1 = (const float*)d_in[8];
  const float* a1 = (const float*)d_in[9];
  const float* W2 = (const float*)d_in[10];
  const float* b2 = (const float*)d_in[11];
  const float* a2 = (const float*)d_in[12];
  const float* Wd = (const float*)d_in[13];
  const float* bd = (const float*)d_in[14];
  float* out = (float*)d_out;
  float* ws  = (float*)d_ws;

  size_t o = 0;
  float* fts   = ws + o; o += (size_t)NN * NH;   // reused for all 4 feature GEMMs
  float* h1    = ws + o; o += (size_t)NN * NH;
  float* h2    = ws + o; o += (size_t)NN * NH;
  float* h3    = ws + o; o += (size_t)NN * NH;
  float* h4    = ws + o; o += (size_t)NN * NH;
  float* Z     = ws + o; o += (size_t)NN * NH;
  float* Zn    = ws + o; o += (size_t)NN * NH;
  float* Zt    = ws + o; o += (size_t)NH * NN;
  float* simf  = ws + o; o += (size_t)NH * NH;
  float* cn    = ws + o; o += NH;
  float* c1    = ws + o; o += NH;
  float* c2    = ws + o; o += NH;
  float* u1    = ws + o; o += NH;
  float* u2    = ws + o; o += NH;
  float* fterm = ws + o; o += NH;
  float* nterm = ws + o; o += NN;
  float* simn  = ws + o; o += (size_t)NN * NN;   // 64 MB

  dim3 blk(256);
  dim3 gNH(NH / 64, NN / 128);     // [4096 x 512] outputs

  // 4x GCN: fts = seq @ W^T, then h = prelu(adj_or_diff @ fts + b)
  hipLaunchKernelGGL((gemm_wmma_f32<1, 0>), gNH, blk, 0, stream, seq1, W1, fts, NN, NH, NH, (const float*)nullptr, (const float*)nullptr);
  hipLaunchKernelGGL((gemm_wmma_f32<0, 1>), gNH, blk, 0, stream, adj,  fts, h1,  NN, NH, NN, b1, a1);
  hipLaunchKernelGGL((gemm_wmma_f32<1, 0>), gNH, blk, 0, stream, seq1, W2, fts, NN, NH, NH, (const float*)nullptr, (const float*)nullptr);
  hipLaunchKernelGGL((gemm_wmma_f32<0, 1>), gNH, blk, 0, stream, diff, fts, h2,  NN, NH, NN, b2, a2);
  hipLaunchKernelGGL((gemm_wmma_f32<1, 0>), gNH, blk, 0, stream, seq2, W1, fts, NN, NH, NH, (const float*)nullptr, (const float*)nullptr);
  hipLaunchKernelGGL((gemm_wmma_f32<0, 1>), gNH, blk, 0, stream, adj,  fts, h3,  NN, NH, NN, b1, a1);
  hipLaunchKernelGGL((gemm_wmma_f32<1, 0>), gNH, blk, 0, stream, seq2, W2, fts, NN, NH, NH, (const float*)nullptr, (const float*)nullptr);
  hipLaunchKernelGGL((gemm_wmma_f32<0, 1>), gNH, blk, 0, stream, diff, fts, h4,  NN, NH, NN, b2, a2);

  // discriminator path
  colmean_sigmoid<<<2, 256, 0, stream>>>(h1, c1);
  colmean_sigmoid<<<2, 256, 0, stream>>>(h2, c2);
  gemv512<<<64, 256, 0, stream>>>(Wd, c1, u1);
  gemv512<<<64, 256, 0, stream>>>(Wd, c2, u2);
  disc_kernel<<<512, 256, 0, stream>>>(h1, h2, h3, h4, u1, u2, bd, out);

  // Z and normalizations
  zadd_kernel<<<(NN * NH) / 256, 256, 0, stream>>>(h1, h2, Z, NN * NH);
  colnorm_kernel<<<2, 256, 0, stream>>>(Z, cn);
  rownorm_normalize<<<NN, 256, 0, stream>>>(Z, Zn);
  transpose_coln<<<dim3(NH / 32, NN / 32), 256, 0, stream>>>(Z, cn, Zt);

  // similarity matrices with fused exp(2x^2) epilogue
  hipLaunchKernelGGL((gemm_wmma_f32<1, 2>), dim3(NH / 64, NH / 128), blk, 0, stream, Zt, Zt, simf, NH, NH, NN, (const float*)nullptr, (const float*)nullptr);
  hipLaunchKernelGGL((gemm_wmma_f32<1, 2>), dim3(NN / 64, NN / 128), blk, 0, stream, Zn, Zn, simn, NN, NN, NH, (const float*)nullptr, (const float*)nullptr);

  // self-paced contrastive losses
  featloss_cols<<<NH, 256, 0, stream>>>(simf, fidx, fterm);
  nodeloss_rows<<<NN, 256, 0, stream>>>(simn, albl, nidx, nterm);
  finalize_losses<<<1, 256, 0, stream>>>(fterm, nterm, out);
}